// DirectedEncoder_9534827397139
// MI455X (gfx1250) — compile-verified
//
#include <hip/hip_runtime.h>

typedef float v2f __attribute__((ext_vector_type(2)));
typedef float v8f __attribute__((ext_vector_type(8)));

#define DIN 128
#define DOUT 128

// ---------------------------------------------------------------------------
// Zero both aggregation buffers (agg_sd in ws, agg_ds aliased on d_out).
// ---------------------------------------------------------------------------
__global__ void de_zero_kernel(float4* __restrict__ a, float4* __restrict__ b, long n4) {
    long i = (long)blockIdx.x * blockDim.x + threadIdx.x;
    long stride = (long)gridDim.x * blockDim.x;
    float4 z; z.x = 0.f; z.y = 0.f; z.z = 0.f; z.w = 0.f;
    for (; i < n4; i += stride) { a[i] = z; b[i] = z; }
}

// ---------------------------------------------------------------------------
// Pack 0.5*[W_sd | W_ds] (K=256) into WMMA B-fragment order, and fold biases.
// B[k][n] = 0.5 * W'[n][k].  Fragment (kt,nt): lane l<16 holds k=4kt+{0,1},
// lane l>=16 holds k=4kt+{2,3}, col n = 16*nt + (l&15).   Flat float index:
//   i = ((kt*8 + nt)*32 + lane)*2 + v
// ---------------------------------------------------------------------------
__global__ void de_pack_kernel(const float* __restrict__ W_sd, const float* __restrict__ W_ds,
                               const float* __restrict__ b_sd, const float* __restrict__ b_ds,
                               float* __restrict__ Bpack, float* __restrict__ biasc) {
    int i = blockIdx.x * blockDim.x + threadIdx.x;
    if (i < 64 * 8 * 32 * 2) {
        int v    = i & 1;
        int lane = (i >> 1) & 31;
        int nt   = (i >> 6) & 7;
        int kt   = i >> 9;
        int col  = nt * 16 + (lane & 15);
        int k    = kt * 4 + 2 * (lane >> 4) + v;
        float val = (k < DIN) ? W_sd[col * DIN + k] : W_ds[col * DIN + (k - DIN)];
        Bpack[i] = 0.5f * val;
    }
    if (i < DOUT) biasc[i] = 0.5f * (b_sd[i] + b_ds[i]);
}

// ---------------------------------------------------------------------------
// Edge aggregation: one wave32 per edge.  Each lane handles a float4 (4 cols).
// agg_sd[dst] += x[src];  agg_ds[src] += x[dst].
// unsafeAtomicAdd -> native non-returning GLOBAL_ATOMIC_ADD_F32 resolving in
// L2 (x + both agg buffers = ~153MB, resident in the 192MB L2).
// Edge indices are single-use: stream them with non-temporal loads so they
// don't evict the hot feature/agg working set.
// ---------------------------------------------------------------------------
__global__ __launch_bounds__(256)
void de_agg_kernel(const float* __restrict__ x,
                   const int* __restrict__ esrc, const int* __restrict__ edst,
                   float* __restrict__ agg_sd, float* __restrict__ agg_ds, int nedges) {
    int e    = blockIdx.x * 8 + (threadIdx.x >> 5);
    int lane = threadIdx.x & 31;
    if (e >= nedges) return;
    int s = __builtin_nontemporal_load(esrc + e);
    int d = __builtin_nontemporal_load(edst + e);
    float4 xs = ((const float4*)(x + (size_t)s * DIN))[lane];
    float4 xd = ((const float4*)(x + (size_t)d * DIN))[lane];
    float* psd = agg_sd + (size_t)d * DIN + lane * 4;
    float* pds = agg_ds + (size_t)s * DIN + lane * 4;
    unsafeAtomicAdd(psd + 0, xs.x); unsafeAtomicAdd(psd + 1, xs.y);
    unsafeAtomicAdd(psd + 2, xs.z); unsafeAtomicAdd(psd + 3, xs.w);
    unsafeAtomicAdd(pds + 0, xd.x); unsafeAtomicAdd(pds + 1, xd.y);
    unsafeAtomicAdd(pds + 2, xd.z); unsafeAtomicAdd(pds + 3, xd.w);
}

// ---------------------------------------------------------------------------
// Fused GEMM: out[n,:] = A'[n,:] (K=256: agg_sd|agg_ds) @ Bpack + biasc
// One wave -> 16 rows x 128 cols (8 x v8f accumulators), 512 WMMAs per wave.
// Block = 256 threads = 8 waves -> 128 rows per block.
// agg_ds aliases `out`; each wave reads only its own rows before writing them,
// so the in-place transform is race-free for all live outputs.  Final stores
// are non-temporal (written once, never re-read).
// ---------------------------------------------------------------------------
__global__ __launch_bounds__(256)
void de_gemm_kernel(const float* __restrict__ agg_sd, const float* __restrict__ agg_ds,
                    const float* __restrict__ Bpack, const float* __restrict__ biasc,
                    float* __restrict__ out, int nrows) {
    const int lane    = threadIdx.x & 31;
    const int wave    = threadIdx.x >> 5;
    const int lm      = lane & 15;
    const int half    = lane >> 4;
    const int rowBase = blockIdx.x * 128 + wave * 16;

    // A-fragment row for this lane (clamped so tail lanes stay in bounds;
    // their results are discarded by the predicated store below).
    int ar = rowBase + lm;
    if (ar >= nrows) ar = nrows - 1;
    const float* aSd = agg_sd + (size_t)ar * DIN;
    const float* aDs = agg_ds + (size_t)ar * DIN;

    // Hoist bias so the epilogue doesn't stall on loads.
    float bv[8];
#pragma unroll
    for (int nt = 0; nt < 8; ++nt) bv[nt] = biasc[nt * 16 + lm];

    v8f acc[8] = {};
    const v2f* Bf = (const v2f*)Bpack;
    const int koff0 = 2 * half;

#pragma unroll 4
    for (int kt = 0; kt < 64; ++kt) {
        const float* Abase = (kt < 32) ? aSd : aDs;      // uniform per kt
        int koff = ((kt & 31) << 2) + koff0;
        v2f a = *(const v2f*)(Abase + koff);
#pragma unroll
        for (int nt = 0; nt < 8; ++nt) {
            v2f b = Bf[(size_t)((kt << 3) + nt) * 32 + lane];
            acc[nt] = __builtin_amdgcn_wmma_f32_16x16x4_f32(
                false, a, false, b, (short)0, acc[nt], false, false);
        }
    }

    // C/D layout: VGPR r -> M = r (lanes 0-15) or M = 8+r (lanes 16-31), N = lm.
#pragma unroll
    for (int nt = 0; nt < 8; ++nt) {
#pragma unroll
        for (int r = 0; r < 8; ++r) {
            int row = rowBase + r + 8 * half;
            if (row < nrows)
                __builtin_nontemporal_store(acc[nt][r] + bv[nt],
                                            out + (size_t)row * DOUT + nt * 16 + lm);
        }
    }
}

// ---------------------------------------------------------------------------
extern "C" void kernel_launch(void* const* d_in, const int* in_sizes, int n_in,
                              void* d_out, int out_size, void* d_ws, size_t ws_size,
                              hipStream_t stream) {
    const float* x    = (const float*)d_in[0];
    const int*   esrc = (const int*)d_in[1];
    const int*   edst = (const int*)d_in[2];
    const float* W_sd = (const float*)d_in[3];
    const float* b_sd = (const float*)d_in[4];
    const float* W_ds = (const float*)d_in[5];
    const float* b_ds = (const float*)d_in[6];

    const int N = in_sizes[0] / DIN;   // 100000
    const int E = in_sizes[1];         // 1000000

    float* out    = (float*)d_out;
    float* agg_sd = (float*)d_ws;                      // N*128 floats
    float* Bpack  = agg_sd + (size_t)N * DIN;          // 32768 floats
    float* biasc  = Bpack + 64 * 8 * 32 * 2;           // 128 floats
    float* agg_ds = out;                               // alias onto output

    // 1) zero aggregation buffers
    long n4 = (long)N * (DIN / 4);
    de_zero_kernel<<<1024, 256, 0, stream>>>((float4*)agg_sd, (float4*)agg_ds, n4);

    // 2) pack weights/bias into WMMA fragment order
    de_pack_kernel<<<128, 256, 0, stream>>>(W_sd, W_ds, b_sd, b_ds, Bpack, biasc);

    // 3) edge scatter-add (one wave per edge, native f32 L2 atomics)
    de_agg_kernel<<<(E + 7) / 8, 256, 0, stream>>>(x, esrc, edst, agg_sd, agg_ds, E);

    // 4) fused dual-linear GEMM on the matrix pipes
    de_gemm_kernel<<<(N + 127) / 128, 256, 0, stream>>>(agg_sd, agg_ds, Bpack, biasc, out, N);
}